// UNI_GCN_17093969838443
// MI455X (gfx1250) — compile-verified
//
#include <hip/hip_runtime.h>
#include <hip/hip_bf16.h>

// ---------------------------------------------------------------------------
// UNI_GCN on MI455X (gfx1250): fp32 WMMA (V_WMMA_F32_16X16X4_F32) for both the
// input projection GEMM and the per-dialogue dense-block graph aggregation,
// plus async global->LDS staging (ASYNCcnt path) for the aggregation operand.
// ---------------------------------------------------------------------------

typedef float v2f __attribute__((ext_vector_type(2)));
typedef float v8f __attribute__((ext_vector_type(8)));
typedef int   v4i __attribute__((ext_vector_type(4)));

#define N_NODES 3000
#define NDIM    1024
#define NHID    512
#define NUMK    4
#define DIA     50
#define NDIA    60
#define XS_LD   (NHID + 4)   // LDS row stride pad: kills 512-stride bank alias

#define AS_GLOBAL __attribute__((address_space(1)))
#define AS_LDS    __attribute__((address_space(3)))

#if defined(__has_builtin)
#if __has_builtin(__builtin_amdgcn_global_load_async_to_lds_b128)
#define HAVE_ASYNC_LDS 1
#endif
#endif

// D = A(16x4,f32) * B(4x16,f32) + C(16x16,f32), exact fp32 semantics.
__device__ __forceinline__ v8f wmma_f32_k4(v2f a, v2f b, v8f c) {
    return __builtin_amdgcn_wmma_f32_16x16x4_f32(
        /*neg_a=*/false, a, /*neg_b=*/false, b,
        /*c_mod=*/(short)0, c, /*reuse_a=*/false, /*reuse_b=*/false);
}

__device__ __forceinline__ void wait_async_zero() {
#if defined(__has_builtin)
#if __has_builtin(__builtin_amdgcn_s_wait_asynccnt)
    __builtin_amdgcn_s_wait_asynccnt(0);
    return;
#endif
#endif
    asm volatile("s_wait_asynccnt 0x0" ::: "memory");
}

// ---------------------------------------------------------------------------
// x1 = E @ W1 + b1.  One wave computes a 16x64 output panel (4 accumulators).
// grid.x = 188 row tiles of 16; 8 waves/block cover the 512 columns.
// ---------------------------------------------------------------------------
__global__ __launch_bounds__(256) void gemm_x1_kernel(
    const float* __restrict__ E, const float* __restrict__ W1,
    const float* __restrict__ b1, float* __restrict__ out,
    float* __restrict__ g0)
{
    const int lane = threadIdx.x & 31;
    const int wave = threadIdx.x >> 5;            // 0..7 -> 64-col panel
    const int rowTile = blockIdx.x;               // 0..187
    const int colBase = wave * 64;

    const int m15 = lane & 15;
    const int hi  = lane >> 4;                    // 0 / 1
    const int kA  = hi * 2;                       // A frag K sub-offset (0 or 2)

    const int gm  = rowTile * 16 + m15;
    const int gmc = gm < N_NODES ? gm : (N_NODES - 1);
    const float mval = gm < N_NODES ? 1.0f : 0.0f;   // branchless row pad

    const float* __restrict__ Arow = E + (size_t)gmc * NDIM;

    v8f acc[4];
    acc[0] = v8f{}; acc[1] = v8f{}; acc[2] = v8f{}; acc[3] = v8f{};

    for (int k0 = 0; k0 < NDIM; k0 += 4) {
        v2f a;
        a.x = Arow[k0 + kA]     * mval;
        a.y = Arow[k0 + kA + 1] * mval;
        const float* __restrict__ Bp =
            W1 + (size_t)(k0 + kA) * NHID + colBase + m15;
        #pragma unroll
        for (int t = 0; t < 4; ++t) {
            v2f b;
            b.x = Bp[t * 16];
            b.y = Bp[NHID + t * 16];
            acc[t] = wmma_f32_k4(a, b, acc[t]);
        }
    }

    #pragma unroll
    for (int v = 0; v < 8; ++v) {
        const int mrow = rowTile * 16 + v + hi * 8;
        if (mrow < N_NODES) {
            #pragma unroll
            for (int t = 0; t < 4; ++t) {
                const int col = colBase + t * 16 + m15;
                const float val = acc[t][v] + b1[col];
                out[(size_t)mrow * (2 * NHID) + col] = val;  // x1 half
                g0 [(size_t)mrow * NHID + col]       = val;  // gnn state seed
            }
        }
    }
}

// ---------------------------------------------------------------------------
// degree / dis helpers (generic over edge_index).
// ---------------------------------------------------------------------------
__global__ void zero_f_kernel(float* __restrict__ p, int n) {
    int i = blockIdx.x * blockDim.x + threadIdx.x;
    if (i < n) p[i] = 0.0f;
}
__global__ void count_deg_kernel(const int* __restrict__ col, int nE,
                                 float* __restrict__ deg) {
    int e = blockIdx.x * blockDim.x + threadIdx.x;
    if (e < nE) atomicAdd(&deg[col[e]], 1.0f);
}
__global__ void make_dis_kernel(float* __restrict__ deg, int n) {
    int i = blockIdx.x * blockDim.x + threadIdx.x;
    if (i < n) deg[i] = rsqrtf(fmaxf(deg[i], 1.0f));
}

// ---------------------------------------------------------------------------
// Per-node gate dot products: s_top[n] = x[n].gw[:H], s_bot[n] = x[n].gw[H:].
// One wave per node, shuffle reduction.
// ---------------------------------------------------------------------------
__global__ __launch_bounds__(256) void dots_kernel(
    const float* __restrict__ X, const float* __restrict__ gate_w, int kk,
    float* __restrict__ s_top, float* __restrict__ s_bot)
{
    const int lane = threadIdx.x & 31;
    const int node = blockIdx.x * 8 + (threadIdx.x >> 5);
    if (node >= N_NODES) return;
    const float* __restrict__ x    = X + (size_t)node * NHID;
    const float* __restrict__ gtop = gate_w + (size_t)kk * (2 * NHID);
    const float* __restrict__ gbot = gtop + NHID;
    float st = 0.0f, sb = 0.0f;
    for (int i = 0; i < NHID; i += 32) {
        const float xv = x[i + lane];
        st += xv * gtop[i + lane];
        sb += xv * gbot[i + lane];
    }
    #pragma unroll
    for (int off = 16; off; off >>= 1) {
        st += __shfl_down(st, off, 32);
        sb += __shfl_down(sb, off, 32);
    }
    if (lane == 0) { s_top[node] = st; s_bot[node] = sb; }
}

// ---------------------------------------------------------------------------
// One workgroup per dialogue.
//   1) Kick off async DMA of the 50x512 X block into LDS (ASYNCcnt path).
//   2) Overlap: build A[c,r] = dis[c]*dis[r]*tanh(st[c]+sb[r]+gb) in LDS
//      (zero diagonal, zero-padded to 64x52).
//   3) s_wait_asynccnt 0 + barrier, then Y = X + A @ X via WMMA with both
//      operands sourced from LDS. 8 waves sweep 4 M-tiles x 32 N-tiles.
// ---------------------------------------------------------------------------
__global__ __launch_bounds__(256) void agg_kernel(
    const float* __restrict__ X,        // g_cur  [N, NHID]
    float* __restrict__ Y,              // g_next [N, NHID]
    const float* __restrict__ s_top, const float* __restrict__ s_bot,
    const float* __restrict__ dis, const float* __restrict__ gate_b, int kk)
{
    __shared__ float As[64 * 56];       // coefficient matrix, lda = 56
    __shared__ float Xs[DIA * XS_LD];   // staged X block, 50 rows x 516 stride
    const int off = blockIdx.x * DIA;
    const float gb = gate_b[kk];

    // --- 1) async global->LDS staging of X block (16B chunks) -------------
    const float* __restrict__ srcBase = X + (size_t)off * NHID;
#ifdef HAVE_ASYNC_LDS
    for (int c4 = threadIdx.x; c4 < (DIA * NHID) / 4; c4 += 256) {
        const int row = c4 >> 7;                 // NHID/4 = 128 chunks per row
        const int col = (c4 & 127) * 4;
        __builtin_amdgcn_global_load_async_to_lds_b128(
            (AS_GLOBAL v4i*)(srcBase + row * NHID + col),
            (AS_LDS v4i*)&Xs[row * XS_LD + col], 0, 0);
    }
#else
    for (int c4 = threadIdx.x; c4 < (DIA * NHID) / 4; c4 += 256) {
        const int row = c4 >> 7;
        const int col = (c4 & 127) * 4;
        *(float4*)&Xs[row * XS_LD + col] =
            *(const float4*)(srcBase + row * NHID + col);
    }
#endif

    // --- 2) build coefficient matrix while DMA is in flight ----------------
    for (int i = threadIdx.x; i < 64 * 52; i += 256) {
        const int c = i / 52, r = i % 52;
        const int cc = c < DIA ? c : 0;
        const int rc = r < DIA ? r : 0;
        const float v = dis[off + cc] * dis[off + rc] *
                        tanhf(s_top[off + cc] + s_bot[off + rc] + gb);
        const bool valid = (c < DIA) && (r < DIA) && (r != c);
        As[c * 56 + r] = valid ? v : 0.0f;
    }

#ifdef HAVE_ASYNC_LDS
    wait_async_zero();                  // this wave's DMA complete
#endif
    __syncthreads();                    // all waves' As + Xs visible

    // --- 3) Y = X + A @ X via WMMA f32 16x16x4 -----------------------------
    const int lane = threadIdx.x & 31;
    const int wave = threadIdx.x >> 5;
    const int m15  = lane & 15;
    const int hi   = lane >> 4;
    const int kf   = hi * 2;

    for (int tile = wave; tile < 128; tile += 8) {
        const int mtile = tile >> 5;     // 0..3
        const int ntile = tile & 31;     // 0..31
        const int nb = ntile * 16 + m15;
        v8f acc = v8f{};
        for (int k0 = 0; k0 < 52; k0 += 4) {
            v2f a;
            a.x = As[(mtile * 16 + m15) * 56 + k0 + kf];
            a.y = As[(mtile * 16 + m15) * 56 + k0 + kf + 1];
            const int r0 = k0 + kf, r1 = r0 + 1;
            const int rr0 = r0 < DIA ? r0 : 0;
            const int rr1 = r1 < DIA ? r1 : 0;
            v2f b;
            const float b0 = Xs[rr0 * XS_LD + nb];
            const float b1 = Xs[rr1 * XS_LD + nb];
            b.x = r0 < DIA ? b0 : 0.0f;  // pad rows: A is zero there anyway
            b.y = r1 < DIA ? b1 : 0.0f;
            acc = wmma_f32_k4(a, b, acc);
        }
        #pragma unroll
        for (int v = 0; v < 8; ++v) {
            const int mrow = mtile * 16 + v + hi * 8;
            if (mrow < DIA) {
                Y[(size_t)(off + mrow) * NHID + nb] =
                    Xs[mrow * XS_LD + nb] + acc[v];   // residual add fused
            }
        }
    }
}

// ---------------------------------------------------------------------------
// Final: out[:, NHID:2*NHID] = gnn_out
// ---------------------------------------------------------------------------
__global__ void copy_half_kernel(const float* __restrict__ g,
                                 float* __restrict__ out) {
    int i = blockIdx.x * blockDim.x + threadIdx.x;
    if (i < N_NODES * NHID) {
        const int n = i / NHID, h = i % NHID;
        out[(size_t)n * (2 * NHID) + NHID + h] = g[i];
    }
}

extern "C" void kernel_launch(void* const* d_in, const int* in_sizes, int n_in,
                              void* d_out, int out_size, void* d_ws, size_t ws_size,
                              hipStream_t stream) {
    const float* E    = (const float*)d_in[0];
    // d_in[1] qmask : unused by the reference output path
    const int*   edge = (const int*)d_in[2];
    // d_in[3] epoch : unused
    const float* W1   = (const float*)d_in[4];
    const float* b1   = (const float*)d_in[5];
    const float* gw   = (const float*)d_in[6];
    const float* gb   = (const float*)d_in[7];
    float* out = (float*)d_out;

    float* ws  = (float*)d_ws;
    float* g0  = ws;
    float* g1  = g0 + (size_t)N_NODES * NHID;
    float* stp = g1 + (size_t)N_NODES * NHID;
    float* sbt = stp + N_NODES;
    float* dis = sbt + N_NODES;

    const int nE = in_sizes[2] / 2;               // edge_index is [2, E]
    const int* colIdx = edge + nE;                // dst indices

    // x1 = E @ W1 + b1  (WMMA f32); also seeds GNN state.
    gemm_x1_kernel<<<(N_NODES + 15) / 16, 256, 0, stream>>>(E, W1, b1, out, g0);

    // dis = clamp(deg,1)^(-1/2)
    zero_f_kernel<<<(N_NODES + 255) / 256, 256, 0, stream>>>(dis, N_NODES);
    count_deg_kernel<<<(nE + 255) / 256, 256, 0, stream>>>(colIdx, nE, dis);
    make_dis_kernel<<<(N_NODES + 255) / 256, 256, 0, stream>>>(dis, N_NODES);

    float* cur = g0;
    float* nxt = g1;
    for (int kk = 0; kk < NUMK; ++kk) {
        dots_kernel<<<(N_NODES + 7) / 8, 256, 0, stream>>>(cur, gw, kk, stp, sbt);
        agg_kernel<<<NDIA, 256, 0, stream>>>(cur, nxt, stp, sbt, dis, gb, kk);
        float* t = cur; cur = nxt; nxt = t;
    }

    copy_half_kernel<<<(N_NODES * NHID + 255) / 256, 256, 0, stream>>>(cur, out);
}